// InvPrefImplicit_21363167331017
// MI455X (gfx1250) — compile-verified
//
#include <hip/hip_runtime.h>
#include <math.h>

// ---------------------------------------------------------------------------
// InvPref / LightGCN forward for MI455X (gfx1250, wave32).
//
//  Phase A (x2, inv & env): cur = concat(u,i); acc = cur
//     layer loop x2:  nxt = A*cur  (edge-parallel SpMM, f32 global atomics,
//                                   features L2-resident: 153.6MB < 192MB L2)
//                     acc += nxt ; swap(cur,nxt)
//     gather batch rows of acc/3 into ue_*/ie_* [B x 64]
//  Phase B: fused score kernel — products, sigmoid dot-scores, then the
//     classifier GEMM [16x64]@[64x16(pad)] via chained V_WMMA_F32_16X16X4_F32
//     and a 4-wide log-softmax.
// ---------------------------------------------------------------------------

#define FACTOR 64

typedef __attribute__((ext_vector_type(2))) float v2f;
typedef __attribute__((ext_vector_type(8))) float v8f;

__device__ __forceinline__ float sigmoidf(float x) {
  return 1.0f / (1.0f + __expf(-x));
}

// cur = acc = concat(user_emb, item_emb)   (float4-vectorized streaming)
__global__ __launch_bounds__(256) void concat_init_kernel(
    const float4* __restrict__ u, const float4* __restrict__ it,
    float4* __restrict__ cur, float4* __restrict__ acc,
    long long u4, long long total4) {
  long long idx = (long long)blockIdx.x * blockDim.x + threadIdx.x;
  if (idx >= total4) return;
  float4 v = (idx < u4) ? u[idx] : it[idx - u4];
  cur[idx] = v;
  acc[idx] = v;
}

// One wave32 per edge: lanes cover the 64-wide feature row as float2 each.
// 256B coalesced gather from cur[col], 2 x global_atomic_add_f32 into nxt[row].
__global__ __launch_bounds__(256) void spmm_edge_kernel(
    const int* __restrict__ rows, const int* __restrict__ cols,
    const float* __restrict__ vals, const float* __restrict__ cur,
    float* __restrict__ nxt, int nnz) {
  int wave = blockIdx.x * (blockDim.x >> 5) + (threadIdx.x >> 5);
  if (wave >= nnz) return;
  int lane = threadIdx.x & 31;
  int r = rows[wave];
  int c = cols[wave];
  float v = vals[wave];
  float2 a = *reinterpret_cast<const float2*>(cur + (size_t)c * FACTOR + lane * 2);
  float* dst = nxt + (size_t)r * FACTOR + lane * 2;
  unsafeAtomicAdd(dst, v * a.x);         // -> global_atomic_add_f32 (L2-resident)
  unsafeAtomicAdd(dst + 1, v * a.y);
}

__global__ __launch_bounds__(256) void acc_add_kernel(
    float4* __restrict__ acc, const float4* __restrict__ nxt, long long total4) {
  long long idx = (long long)blockIdx.x * blockDim.x + threadIdx.x;
  if (idx >= total4) return;
  float4 a = acc[idx];
  float4 n = nxt[idx];
  a.x += n.x; a.y += n.y; a.z += n.z; a.w += n.w;
  acc[idx] = a;
}

// ue[b][f] = acc[users_id[b]][f] / 3 ;  ie[b][f] = acc[U + items_id[b]][f] / 3
__global__ __launch_bounds__(256) void gather_scale_kernel(
    const float* __restrict__ acc, const int* __restrict__ uid,
    const int* __restrict__ iid, float* __restrict__ ue, float* __restrict__ ie,
    int batch, int u_num) {
  int idx = blockIdx.x * blockDim.x + threadIdx.x;
  if (idx >= batch * FACTOR) return;
  int b = idx >> 6, f = idx & 63;
  const float s = 1.0f / 3.0f;  // /(N_LAYERS+1)
  ue[idx] = acc[(size_t)uid[b] * FACTOR + f] * s;
  ie[idx] = acc[((size_t)u_num + iid[b]) * FACTOR + f] * s;
}

// 4 waves / block, each wave owns a 16-row batch tile.
// LDS row stride 68 floats => A-frag reads touch 64 distinct banks.
__global__ __launch_bounds__(128) void score_logits_wmma_kernel(
    const float* __restrict__ ue_inv, const float* __restrict__ ie_inv,
    const float* __restrict__ ue_env, const float* __restrict__ ie_env,
    const float* __restrict__ env_emb, const int* __restrict__ envs_id,
    const float* __restrict__ clf_W, const float* __restrict__ clf_b,
    float* __restrict__ out, int batch, int env_num) {
  __shared__ float lds_ip[4][16][68];   // inv_pref tiles (padded)
  __shared__ float lds_lg[4][16][17];   // logits tiles (padded)
  __shared__ float lds_W[16][FACTOR];   // W padded to 16 columns (N dim)

  const int w = threadIdx.x >> 5;
  const int lane = threadIdx.x & 31;
  const int base = (blockIdx.x * 4 + w) * 16;

  for (int t = threadIdx.x; t < 16 * FACTOR; t += 128) {
    int n = t >> 6, k = t & 63;
    lds_W[n][k] = (n < env_num) ? clf_W[n * FACTOR + k] : 0.0f;
  }
  __syncthreads();

  const int f = lane * 2;
  for (int r = 0; r < 16; ++r) {
    const int row = base + r;
    float ip0 = 0.0f, ip1 = 0.0f, ep = 0.0f;
    if (row < batch) {
      float2 ui = *(const float2*)(ue_inv + (size_t)row * FACTOR + f);
      float2 ii = *(const float2*)(ie_inv + (size_t)row * FACTOR + f);
      ip0 = ui.x * ii.x;
      ip1 = ui.y * ii.y;
      float2 u2 = *(const float2*)(ue_env + (size_t)row * FACTOR + f);
      float2 i2 = *(const float2*)(ie_env + (size_t)row * FACTOR + f);
      float2 ee = *(const float2*)(env_emb + (size_t)envs_id[row] * FACTOR + f);
      ep = u2.x * i2.x * ee.x + u2.y * i2.y * ee.y;
    }
    lds_ip[w][r][f] = ip0;
    lds_ip[w][r][f + 1] = ip1;
    float si = ip0 + ip1;
#pragma unroll
    for (int off = 16; off >= 1; off >>= 1) {  // wave32 butterfly reduce
      si += __shfl_xor(si, off, 32);
      ep += __shfl_xor(ep, off, 32);
    }
    if (lane == 0 && row < batch) {
      float isc = sigmoidf(si);
      out[row] = isc;                           // invariant_score
      out[batch + row] = isc * sigmoidf(ep);    // env_aware_score
    }
  }
  __syncthreads();

  // logits tile = inv_pref(16x64) @ B(64x16), K split into 16 x (K=4) WMMAs.
  // 32-bit A 16x4 layout: lanes0-15 M=0..15, VGPR0:K=0|2, VGPR1:K=1|3.
  const int m = lane & 15;
  const int hi = lane >> 4;
  v8f acc = {0.f, 0.f, 0.f, 0.f, 0.f, 0.f, 0.f, 0.f};
#pragma unroll
  for (int k16 = 0; k16 < 16; ++k16) {
    const int k = k16 * 4 + 2 * hi;
    v2f a, b;
    a.x = lds_ip[w][m][k];
    a.y = lds_ip[w][m][k + 1];
    b.x = lds_W[m][k];      // B[k][n] = W[n][k], n = lane&15
    b.y = lds_W[m][k + 1];
    acc = __builtin_amdgcn_wmma_f32_16x16x4_f32(
        false, a, false, b, (short)0, acc, false, false);
  }
  // D layout: VGPR j -> (M=j, N=lane) / (M=j+8, N=lane-16)
#pragma unroll
  for (int j = 0; j < 8; ++j) lds_lg[w][j + 8 * hi][m] = acc[j];
  __syncthreads();

  if (lane < 16) {
    const int row = base + lane;
    if (row < batch) {
      float l[16];
      float mx = -3.402823466e+38f;
      for (int e = 0; e < env_num; ++e) {
        l[e] = lds_lg[w][lane][e] + clf_b[e];
        mx = fmaxf(mx, l[e]);
      }
      float s = 0.0f;
      for (int e = 0; e < env_num; ++e) s += __expf(l[e] - mx);
      float lse = __logf(s) + mx;
      for (int e = 0; e < env_num; ++e)
        out[2 * batch + (size_t)row * env_num + e] = l[e] - lse;  // log_softmax
    }
  }
}

extern "C" void kernel_launch(void* const* d_in, const int* in_sizes, int n_in,
                              void* d_out, int out_size, void* d_ws,
                              size_t ws_size, hipStream_t stream) {
  const int* users_id  = (const int*)d_in[0];
  const int* items_id  = (const int*)d_in[1];
  const int* envs_id   = (const int*)d_in[2];
  // d_in[3] = alpha (identity in forward)
  const int* rows      = (const int*)d_in[4];
  const int* cols      = (const int*)d_in[5];
  const float* vals    = (const float*)d_in[6];
  const float* user_inv = (const float*)d_in[7];
  const float* item_inv = (const float*)d_in[8];
  const float* user_env = (const float*)d_in[9];
  const float* item_env = (const float*)d_in[10];
  const float* env_emb  = (const float*)d_in[11];
  const float* clf_W    = (const float*)d_in[12];
  const float* clf_b    = (const float*)d_in[13];
  float* out = (float*)d_out;

  const int batch  = in_sizes[0];
  const int nnz    = in_sizes[4];
  const int u_num  = in_sizes[7] / FACTOR;
  const int i_num  = in_sizes[8] / FACTOR;
  const int env_num = in_sizes[13];
  const long long NF = (long long)(u_num + i_num) * FACTOR;

  // Workspace layout (floats): 3 node buffers reused across both
  // propagations (~230MB) + 4 gathered batch matrices (~17MB).
  float* bufA = (float*)d_ws;   // cur / nxt ping-pong
  float* bufB = bufA + NF;
  float* bufC = bufB + NF;      // acc
  float* ue_inv = bufC + NF;
  float* ie_inv = ue_inv + (size_t)batch * FACTOR;
  float* ue_env = ie_inv + (size_t)batch * FACTOR;
  float* ie_env = ue_env + (size_t)batch * FACTOR;

  const long long total4 = NF / 4;
  const long long u4 = (long long)u_num * FACTOR / 4;
  const int spmm_blocks = (nnz + 7) / 8;                 // 8 waves / block
  const int ew_blocks = (int)((total4 + 255) / 256);
  const int gat_blocks = (batch * FACTOR + 255) / 256;

  for (int phase = 0; phase < 2; ++phase) {
    const float* ue = phase ? user_env : user_inv;
    const float* ie = phase ? item_env : item_inv;
    float* gu = phase ? ue_env : ue_inv;
    float* gi = phase ? ie_env : ie_inv;

    concat_init_kernel<<<ew_blocks, 256, 0, stream>>>(
        (const float4*)ue, (const float4*)ie, (float4*)bufA, (float4*)bufC,
        u4, total4);

    // layer 1: bufB = A * bufA ; acc += bufB
    hipMemsetAsync(bufB, 0, (size_t)NF * sizeof(float), stream);
    spmm_edge_kernel<<<spmm_blocks, 256, 0, stream>>>(rows, cols, vals, bufA,
                                                      bufB, nnz);
    acc_add_kernel<<<ew_blocks, 256, 0, stream>>>((float4*)bufC,
                                                  (const float4*)bufB, total4);
    // layer 2: bufA = A * bufB ; acc += bufA
    hipMemsetAsync(bufA, 0, (size_t)NF * sizeof(float), stream);
    spmm_edge_kernel<<<spmm_blocks, 256, 0, stream>>>(rows, cols, vals, bufB,
                                                      bufA, nnz);
    acc_add_kernel<<<ew_blocks, 256, 0, stream>>>((float4*)bufC,
                                                  (const float4*)bufA, total4);

    gather_scale_kernel<<<gat_blocks, 256, 0, stream>>>(
        bufC, users_id, items_id, gu, gi, batch, u_num);
  }

  const int score_blocks = (batch + 63) / 64;  // 4 tiles of 16 rows / block
  score_logits_wmma_kernel<<<score_blocks, 128, 0, stream>>>(
      ue_inv, ie_inv, ue_env, ie_env, env_emb, envs_id, clf_W, clf_b, out,
      batch, env_num);
}